// MultiHeadAttention_9620726743529
// MI455X (gfx1250) — compile-verified
//
#include <hip/hip_runtime.h>
#include <hip/hip_bf16.h>

typedef __attribute__((ext_vector_type(2))) float v2f;
typedef __attribute__((ext_vector_type(8))) float v8f;

#define NB 2
#define NS 2048
#define NX 512
#define NH 8
#define ND 64
#define NF 64

__device__ __forceinline__ v8f wmma4(v2f a, v2f b, v8f c) {
  // D = A(16x4 f32) * B(4x16 f32) + C(16x16 f32)
  return __builtin_amdgcn_wmma_f32_16x16x4_f32(false, a, false, b, (short)0, c,
                                               false, false);
}

__device__ __forceinline__ v8f v8zero() {
  v8f z = {0.f, 0.f, 0.f, 0.f, 0.f, 0.f, 0.f, 0.f};
  return z;
}

// ---------------------------------------------------------------------------
// Kernel 1: QKV projection.  out[A,n,b,s,h] = sum_x x[b,s,x]*qkv[x,A,n,h]
// M = B*S = 4096 rows, N = 3*H*D = 1536 cols, K = 512.
// One wave computes one 16x16 tile with 128 chained f32 WMMAs (dual accum).
// Q gets bias and the 1/sqrt(D) scale folded in; stored as [b,n,s,h].
// ---------------------------------------------------------------------------
__global__ void __launch_bounds__(128)
qkv_proj_kernel(const float* __restrict__ x, const float* __restrict__ w,
                const float* __restrict__ qbias, float* __restrict__ Q,
                float* __restrict__ K, float* __restrict__ V) {
  const int lane = threadIdx.x & 31;
  const int wave = threadIdx.x >> 5;
  const int l16 = lane & 15;
  const int half = lane >> 4;
  const int koff = 2 * half;

  const int nt = blockIdx.x;             // 0..95  (N tiles)
  const int mt = blockIdx.y * 4 + wave;  // 0..255 (M tiles)
  const int m0 = mt * 16, n0 = nt * 16;
  const int arow = m0 + l16;

  v8f acc0 = v8zero(), acc1 = v8zero();
#pragma unroll 4
  for (int c = 0; c < 128; c += 2) {
    int kk = 4 * c + koff;
    v2f a0 = *(const v2f*)(x + arow * NX + kk);
    v2f b0;
    b0.x = w[kk * 1536 + n0 + l16];
    b0.y = w[(kk + 1) * 1536 + n0 + l16];
    acc0 = wmma4(a0, b0, acc0);
    int kk1 = kk + 4;
    v2f a1 = *(const v2f*)(x + arow * NX + kk1);
    v2f b1;
    b1.x = w[kk1 * 1536 + n0 + l16];
    b1.y = w[(kk1 + 1) * 1536 + n0 + l16];
    acc1 = wmma4(a1, b1, acc1);
  }
  v8f acc = acc0 + acc1;

  const int col = n0 + l16;
  const int Aidx = col >> 9;  // 0:Q 1:K 2:V
  const int rem = col & 511;
  const int n = rem >> 6;
  const int h = rem & 63;
  const float scale = 0.125f;  // 64^-0.5
  const float bias = (Aidx == 0) ? qbias[n * ND + h] : 0.0f;
  float* dst = (Aidx == 0) ? Q : (Aidx == 1) ? K : V;

#pragma unroll
  for (int v = 0; v < 8; ++v) {
    int m = m0 + v + 8 * half;
    int bb = m >> 11;       // / NS
    int s = m & (NS - 1);
    float val = acc[v];
    if (Aidx == 0) val = (val + bias) * scale;
    dst[(((size_t)bb * NH + n) * NS + s) * ND + h] = val;
  }
}

// ---------------------------------------------------------------------------
// Kernel 2: positional embedding  E[n,v,h] = sum_f sin(freq_f*v+phase_f)*P[f,n,h]
// Tiny (67 MFLOP): plain VALU.
// ---------------------------------------------------------------------------
__global__ void __launch_bounds__(256)
emb_kernel(const float* __restrict__ pos, float* __restrict__ E) {
  const int tid = blockIdx.x * blockDim.x + threadIdx.x;  // n*NS*ND + v*ND + h
  const int h = tid & (ND - 1);
  const int v = (tid >> 6) & (NS - 1);
  const int n = tid >> 17;
  const float PI = 3.14159265358979323846f;
  float acc = 0.f;
#pragma unroll 4
  for (int f = 0; f < NF; ++f) {
    float e = (float)(f >> 1) * (1.0f / 31.0f);  // (f//2)/(NF/2-1)
    float freq = PI * __powf(2.0f / 10000.0f, e);
    float phase = (f & 1) ? (PI * 0.5f) : 0.f;
    float sv = __sinf(freq * (float)v + phase);
    acc += sv * pos[f * (NH * ND) + n * ND + h];
  }
  E[tid] = acc;
}

// ---------------------------------------------------------------------------
// Kernel 3: fused causal relative attention (flash style).
// One wave owns a 16-row query tile; loops over k-tiles of 16.
// Per k-tile: 16 WMMA (QK^T) + 32 WMMA (Q*emb^T strip, 16x32) + 16 WMMA (PV).
// rel gather done with lane shuffles; P transposed to A-layout through LDS.
// ---------------------------------------------------------------------------
__global__ void __launch_bounds__(128)
attn_kernel(const float* __restrict__ Q, const float* __restrict__ K,
            const float* __restrict__ V, const float* __restrict__ E,
            float* __restrict__ O) {
  __shared__ float lds[4][16 * 18];
  const int lane = threadIdx.x & 31;
  const int wave = threadIdx.x >> 5;
  const int l16 = lane & 15;
  const int half = lane >> 4;
  const int koff = 2 * half;

  const int wid = blockIdx.x * 4 + wave;  // 0..2047
  const int qt = wid & 127;
  const int n = (wid >> 7) & 7;
  const int b = wid >> 10;
  const int q0 = qt * 16;

  const float* Qp = Q + ((size_t)(b * NH + n) * NS) * ND;
  const float* Kp = K + ((size_t)(b * NH + n) * NS) * ND;
  const float* Vp = V + ((size_t)(b * NH + n) * NS) * ND;
  const float* Ep = E + (size_t)n * NS * ND;

  // Q tile kept resident as 16 A-operand chunks (32 floats/lane)
  v2f qa[16];
#pragma unroll
  for (int c = 0; c < 16; ++c)
    qa[c] = *(const v2f*)(Qp + (q0 + l16) * ND + 4 * c + koff);

  v8f oacc[4] = {v8zero(), v8zero(), v8zero(), v8zero()};
  float mrow[8], lrow[8];
#pragma unroll
  for (int v = 0; v < 8; ++v) { mrow[v] = -3.0e38f; lrow[v] = 0.f; }

  for (int kt = 0; kt <= qt; ++kt) {
    const int k0 = kt * 16;

    // ---- scores = Q * K^T (already scaled via Q) ----
    v8f s = v8zero();
#pragma unroll
    for (int c = 0; c < 16; ++c) {
      v2f kb = *(const v2f*)(Kp + (k0 + l16) * ND + 4 * c + koff);
      s = wmma4(qa[c], kb, s);
    }

    // ---- relative strip T[16 x 32], columns j -> v = (q0-k0-15)+j ----
    const int vb = q0 - k0 - 15;
    int v1 = vb + l16;       v1 = v1 < 0 ? 0 : (v1 > NS - 1 ? NS - 1 : v1);
    int v2i = vb + 16 + l16; v2i = v2i < 0 ? 0 : (v2i > NS - 1 ? NS - 1 : v2i);
    v8f t1 = v8zero(), t2 = v8zero();
#pragma unroll
    for (int c = 0; c < 16; ++c) {
      v2f e1 = *(const v2f*)(Ep + v1 * ND + 4 * c + koff);
      t1 = wmma4(qa[c], e1, t1);
      v2f e2 = *(const v2f*)(Ep + v2i * ND + 4 * c + koff);
      t2 = wmma4(qa[c], e2, t2);
    }

    // ---- gather rel[q,k] = T[q, 15+q-k]; apply causal mask ----
#pragma unroll
    for (int v = 0; v < 8; ++v) {
      int row = v + 8 * half;
      int idx = 15 + row - l16;              // 0..30
      int src = (lane & 16) | (idx & 15);    // same row, same half
      float r1 = __shfl(t1[v], src, 32);
      float r2 = __shfl(t2[v], src, 32);
      float rel = (idx < 16) ? r1 : r2;
      bool ok = (k0 + l16) <= (q0 + row);
      s[v] = s[v] + (ok ? rel : -1000.0f);
    }

    // ---- online softmax update (rows live in lane halves) ----
    float p[8];
#pragma unroll
    for (int v = 0; v < 8; ++v) {
      float rmax = s[v];
      rmax = fmaxf(rmax, __shfl_xor(rmax, 1, 32));
      rmax = fmaxf(rmax, __shfl_xor(rmax, 2, 32));
      rmax = fmaxf(rmax, __shfl_xor(rmax, 4, 32));
      rmax = fmaxf(rmax, __shfl_xor(rmax, 8, 32));
      float newm = fmaxf(mrow[v], rmax);
      float alpha = __expf(mrow[v] - newm);
      float pv = __expf(s[v] - newm);
      float rsum = pv;
      rsum += __shfl_xor(rsum, 1, 32);
      rsum += __shfl_xor(rsum, 2, 32);
      rsum += __shfl_xor(rsum, 4, 32);
      rsum += __shfl_xor(rsum, 8, 32);
      lrow[v] = lrow[v] * alpha + rsum;
      mrow[v] = newm;
      oacc[0][v] *= alpha; oacc[1][v] *= alpha;
      oacc[2][v] *= alpha; oacc[3][v] *= alpha;
      p[v] = pv;
    }

    // ---- P (D layout) -> A layout via per-wave LDS transpose ----
#pragma unroll
    for (int v = 0; v < 8; ++v)
      lds[wave][(v + 8 * half) * 18 + l16] = p[v];
    asm volatile("s_wait_dscnt 0" ::: "memory");  // stores visible to own wave

    // ---- O += P * V ----
#pragma unroll
    for (int kc = 0; kc < 4; ++kc) {
      // (l16*18 + 4*kc + 2*half)*4B is 8-byte aligned -> single ds_load_b64
      v2f pa;
      pa.x = lds[wave][l16 * 18 + 4 * kc + koff];
      pa.y = lds[wave][l16 * 18 + 4 * kc + koff + 1];
      const int krow = k0 + 4 * kc + koff;
#pragma unroll
      for (int nc = 0; nc < 4; ++nc) {
        v2f vb2;
        vb2.x = Vp[krow * ND + nc * 16 + l16];
        vb2.y = Vp[(krow + 1) * ND + nc * 16 + l16];
        oacc[nc] = wmma4(pa, vb2, oacc[nc]);
      }
    }
    asm volatile("" ::: "memory");  // keep next iter's LDS stores after loads
  }

  // ---- epilogue: O[b, q, n*64+h] = acc / l ----
#pragma unroll
  for (int v = 0; v < 8; ++v) {
    int q = q0 + v + 8 * half;
    float inv = 1.0f / lrow[v];
    float* dst = O + ((size_t)b * NS + q) * (NH * ND) + n * ND;
#pragma unroll
    for (int nc = 0; nc < 4; ++nc) dst[nc * 16 + l16] = oacc[nc][v] * inv;
  }
}

// ---------------------------------------------------------------------------
// Kernel 4: output projection.  out[m, c] = attn[m, :] @ out_w[:, c] + out_b[c]
// M = 4096, N = 512, K = 512.
// ---------------------------------------------------------------------------
__global__ void __launch_bounds__(128)
outproj_kernel(const float* __restrict__ attn, const float* __restrict__ w,
               const float* __restrict__ bias, float* __restrict__ out) {
  const int lane = threadIdx.x & 31;
  const int wave = threadIdx.x >> 5;
  const int l16 = lane & 15;
  const int half = lane >> 4;
  const int koff = 2 * half;

  const int nt = blockIdx.x;             // 0..31
  const int mt = blockIdx.y * 4 + wave;  // 0..255
  const int m0 = mt * 16, n0 = nt * 16;
  const int arow = m0 + l16;

  v8f acc0 = v8zero(), acc1 = v8zero();
#pragma unroll 4
  for (int c = 0; c < 128; c += 2) {
    int kk = 4 * c + koff;
    v2f a0 = *(const v2f*)(attn + arow * NX + kk);
    v2f b0;
    b0.x = w[kk * NX + n0 + l16];
    b0.y = w[(kk + 1) * NX + n0 + l16];
    acc0 = wmma4(a0, b0, acc0);
    int kk1 = kk + 4;
    v2f a1 = *(const v2f*)(attn + arow * NX + kk1);
    v2f b1;
    b1.x = w[kk1 * NX + n0 + l16];
    b1.y = w[(kk1 + 1) * NX + n0 + l16];
    acc1 = wmma4(a1, b1, acc1);
  }
  v8f acc = acc0 + acc1;

  const float bcol = bias[n0 + l16];
#pragma unroll
  for (int v = 0; v < 8; ++v)
    out[(size_t)(m0 + v + 8 * half) * NX + n0 + l16] = acc[v] + bcol;
}

// ---------------------------------------------------------------------------
extern "C" void kernel_launch(void* const* d_in, const int* in_sizes, int n_in,
                              void* d_out, int out_size, void* d_ws,
                              size_t ws_size, hipStream_t stream) {
  const float* x = (const float*)d_in[0];
  const float* qkv = (const float*)d_in[1];
  const float* qbias = (const float*)d_in[2];
  const float* positional = (const float*)d_in[3];
  const float* out_w = (const float*)d_in[4];
  const float* out_b = (const float*)d_in[5];
  float* out = (float*)d_out;

  // workspace: Q,K,V [b,n,s,h] (2M floats each) + emb [n,s,h] (1M) + attn (2M)
  float* ws = (float*)d_ws;
  const size_t qkvsz = (size_t)NB * NH * NS * ND;
  float* Q = ws;
  float* K = Q + qkvsz;
  float* V = K + qkvsz;
  float* E = V + qkvsz;
  float* AT = E + (size_t)NH * NS * ND;

  qkv_proj_kernel<<<dim3(96, 64), 128, 0, stream>>>(x, qkv, qbias, Q, K, V);
  emb_kernel<<<dim3((NH * NS * ND) / 256), 256, 0, stream>>>(positional, E);
  attn_kernel<<<dim3(512), 128, 0, stream>>>(Q, K, V, E, AT);
  outproj_kernel<<<dim3(32, 64), 128, 0, stream>>>(AT, out_w, out_b, out);
}